// GCN_68410239091162
// MI455X (gfx1250) — compile-verified
//
#include <hip/hip_runtime.h>
#include <cstdint>

typedef __attribute__((ext_vector_type(2))) float v2f;
typedef __attribute__((ext_vector_type(8))) float v8f;

// ---------------- utility kernels ----------------

__global__ void zero_f32(float* __restrict__ p, int n) {
    int i = blockIdx.x * blockDim.x + threadIdx.x;
    if (i < n) p[i] = 0.0f;
}

// deg[dst[e]] += 1 for every edge (self-loop added in dinv_kernel)
__global__ void deg_kernel(const long long* __restrict__ dst,
                           float* __restrict__ deg, int E) {
    int e = blockIdx.x * blockDim.x + threadIdx.x;
    if (e < E) atomicAdd(&deg[(int)dst[e]], 1.0f);
}

// deg -> dinv = rsqrt(deg + 1)
__global__ void dinv_kernel(float* __restrict__ deg, int n) {
    int i = blockIdx.x * blockDim.x + threadIdx.x;
    if (i < n) deg[i] = rsqrtf(deg[i] + 1.0f);
}

// ---------------- fp32 WMMA GEMM ----------------
// C[M,N] = op(A)[M,K] x B[K,N], all row-major. M%16==0, N%16==0, K%4==0.
// op(A) = relu(A) when RELU_A (fuses the inter-layer activation into the
// A-fragment loads, eliminating a full pass over the feature matrix).
// One wave per 16x16 output tile. blockDim.x = 32*(N/16), gridDim.x = M/16.
// A-frag layout (V_WMMA_F32_16X16X4_F32): lanes 0-15 hold row M=lane, K=k0..k0+1;
// lanes 16-31 hold row M=lane-16, K=k0+2..k0+3.  B-frag mirrors on K halves.
// D layout: VGPR r -> row (r + 8*(lane>=16)), col lane&15.
template <bool RELU_A>
__global__ void wmma_gemm_f32(const float* __restrict__ A,
                              const float* __restrict__ B,
                              float* __restrict__ C,
                              int M, int K, int N) {
    const int lane  = threadIdx.x & 31;
    const int ntile = threadIdx.x >> 5;
    const int m0    = blockIdx.x * 16;
    const int n0    = ntile * 16;
    const int l15   = lane & 15;          // row of A frag / col of B frag
    const int khalf = lane >> 4;          // 0: K 0-1, 1: K 2-3

    const float* arow = A + (size_t)(m0 + l15) * K;

    v8f acc = {};
    for (int k0 = 0; k0 < K; k0 += 4) {
        const int ka = k0 + khalf * 2;
        v2f a;
        a.x = arow[ka];
        a.y = arow[ka + 1];
        if (RELU_A) {
            a.x = fmaxf(a.x, 0.0f);
            a.y = fmaxf(a.y, 0.0f);
        }
        v2f b;
        b.x = B[(size_t)ka * N + n0 + l15];
        b.y = B[(size_t)(ka + 1) * N + n0 + l15];
        // 8 args: (neg_a, A, neg_b, B, c_mod, C, reuse_a, reuse_b)
        acc = __builtin_amdgcn_wmma_f32_16x16x4_f32(
            false, a, false, b, (short)0, acc, false, false);
    }

    float* crow = C + (size_t)(m0 + 8 * khalf) * N + n0 + l15;
#pragma unroll
    for (int r = 0; r < 8; ++r) {
        crow[(size_t)r * N] = acc[r];
    }
}

// ---------------- accumulator initializer ----------------
// out[i] = h[i] * dinv[node]^2 + bias[f]   (self-loop message + bias).
// Replaces a plain zero-fill at identical traffic cost and deletes the
// post-scatter epilogue pass entirely.
__global__ void init_selfloop(const float* __restrict__ h,
                              const float* __restrict__ dinv,
                              const float* __restrict__ bias,
                              float* __restrict__ out, int total, int F) {
    int i = blockIdx.x * blockDim.x + threadIdx.x;
    if (i >= total) return;
    int node = i / F;
    int f    = i - node * F;
    float di = dinv[node];
    out[i] = h[i] * di * di + bias[f];
}

// ---------------- edge scatter ----------------
// One wave per edge. F==64: float2 per lane; F==32: one float per lane.
// Accumulates directly into the (pre-initialized) destination buffer.
template <int F>
__global__ void scatter_kernel(const long long* __restrict__ src,
                               const long long* __restrict__ dst,
                               const float* __restrict__ dinv,
                               const float* __restrict__ h,
                               float* __restrict__ agg, int E) {
    int warp = (int)((blockIdx.x * blockDim.x + threadIdx.x) >> 5);
    int lane = threadIdx.x & 31;
    if (warp >= E) return;
    int s = (int)src[warp];
    int d = (int)dst[warp];
    float coef = dinv[s] * dinv[d];
    if (F == 64) {
        const float2* hp = (const float2*)(h + (size_t)s * 64);
        float2 v = hp[lane];
        float* ap = agg + (size_t)d * 64 + lane * 2;
        atomicAdd(ap,     v.x * coef);
        atomicAdd(ap + 1, v.y * coef);
    } else {
        float v = h[(size_t)s * F + lane];
        atomicAdd(&agg[(size_t)d * F + lane], v * coef);
    }
}

// ---------------- launch ----------------

extern "C" void kernel_launch(void* const* d_in, const int* in_sizes, int n_in,
                              void* d_out, int out_size, void* d_ws, size_t ws_size,
                              hipStream_t stream) {
    const float*     x   = (const float*)d_in[0];       // [N,128]
    const long long* ei  = (const long long*)d_in[1];   // [2,E] int64
    const float*     W1  = (const float*)d_in[2];       // [128,64]
    const float*     b1  = (const float*)d_in[3];       // [64]
    const float*     W2  = (const float*)d_in[4];       // [64,32]
    const float*     b2  = (const float*)d_in[5];       // [32]

    const int N = in_sizes[0] / 128;   // 100000
    const int E = in_sizes[1] / 2;     // 3200000
    const long long* src = ei;
    const long long* dst = ei + E;

    // workspace layout: dinv[N] | h[N*64] | agg[N*64]
    char*  ws   = (char*)d_ws;
    float* dinv = (float*)ws;
    float* h    = (float*)(ws + (size_t)N * 4);
    float* agg  = (float*)(ws + (size_t)N * 4 + (size_t)N * 64 * 4);
    float* out  = (float*)d_out;

    const int TB = 256;

    // degrees -> dinv
    zero_f32<<<(N + TB - 1) / TB, TB, 0, stream>>>(dinv, N);
    deg_kernel<<<(E + TB - 1) / TB, TB, 0, stream>>>(dst, dinv, E);
    dinv_kernel<<<(N + TB - 1) / TB, TB, 0, stream>>>(dinv, N);

    // ---- layer 1 ----
    // h1 = x @ W1   (M=N, K=128, Nout=64): 4 waves/block, one 16x16 tile each
    wmma_gemm_f32<false><<<N / 16, 32 * 4, 0, stream>>>(x, W1, h, N, 128, 64);
    // agg := h1*dinv^2 + b1  (self-loop + bias as initializer, no zero pass)
    init_selfloop<<<(N * 64 + TB - 1) / TB, TB, 0, stream>>>(h, dinv, b1, agg,
                                                             N * 64, 64);
    // one wave per edge; 8 waves per 256-thread block. agg now holds x1.
    scatter_kernel<64><<<(E + 7) / 8, TB, 0, stream>>>(src, dst, dinv, h, agg, E);

    // ---- layer 2 ----
    // h2 = relu(x1) @ W2  (ReLU fused into A loads); reuse h buffer for h2
    wmma_gemm_f32<true><<<N / 16, 32 * 2, 0, stream>>>(agg, W2, h, N, 64, 32);
    // d_out := h2*dinv^2 + b2, then scatter-add neighbor messages directly in
    init_selfloop<<<(N * 32 + TB - 1) / TB, TB, 0, stream>>>(h, dinv, b2, out,
                                                             N * 32, 32);
    scatter_kernel<32><<<(E + 7) / 8, TB, 0, stream>>>(src, dst, dinv, h, out, E);
}